// BSIGNN_71597104824917
// MI455X (gfx1250) — compile-verified
//
#include <hip/hip_runtime.h>
#include <hip/hip_bf16.h>

// ---------------- Problem constants (from reference setup_inputs) ----------
#define Bb   8
#define Nn   180
#define Tt   256
#define Ll   16
#define HH   64
#define Ss   240   // T - L, exactly 15 WMMA tiles of 16

typedef __attribute__((ext_vector_type(2))) float v2f;
typedef __attribute__((ext_vector_type(8))) float v8f;

// V_WMMA_F32_16X16X4_F32 : D(16x16 f32) = A(16x4 f32) x B(4x16 f32) + C
// A frag (2 VGPR): lane m=l&15, k = kb + 2*(l>>4) {+0,+1}
// B frag (2 VGPR): lane n=l&15, k = kb + 2*(l>>4) {+0,+1}
// C/D: reg r holds D[m = r + 8*(l>>4), n = l&15]
__device__ __forceinline__ v8f wmma4(v2f a, v2f b, v8f c) {
  return __builtin_amdgcn_wmma_f32_16x16x4_f32(false, a, false, b, (short)0, c,
                                               false, false);
}

// ---------------- branchless fast math (CDNA5 TRANS ops) -------------------
__device__ __forceinline__ float fast_rcp(float x) {
  return __builtin_amdgcn_rcpf(x);            // v_rcp_f32
}
__device__ __forceinline__ float fast_tanh(float x) {
#if __has_builtin(__builtin_amdgcn_tanh_f32)
  return __builtin_amdgcn_tanh_f32(x);        // v_tanh_f32 (gfx1250 TRANS)
#else
  // Builtin not wired up on this toolchain: emit the architected gfx1250
  // TRANS op directly. v_nop satisfies the TRANS result-use hazard slot
  // (ISA 7.3: 1 independent op or V_NOP after a trans op) since the
  // compiler cannot schedule across inline asm.
  float r;
  asm("v_tanh_f32 %0, %1\n\tv_nop" : "=v"(r) : "v"(x));
  return r;
#endif
}
__device__ __forceinline__ float fast_sig(float x) {
  // sigmoid(x) = 0.5*tanh(x/2) + 0.5 : single TRANS op + mul + fma
  return fmaf(0.5f, fast_tanh(0.5f * x), 0.5f);
}

// ===========================================================================
// K1: per (b,n): gates GEMM (skip unused forget gate) + LSTM activations +
//     reduce h/x over s  -> hd[b,n,64], rsum[b,n].   One wave per block.
// ===========================================================================
__global__ __launch_bounds__(32) void k_gates_hd(
    const float* __restrict__ x,     // (B,N,T)
    const float* __restrict__ W_ih,  // (N,256,16)
    const float* __restrict__ b_ih,  // (N,256)
    const float* __restrict__ b_hh,  // (N,256)
    float* __restrict__ hd,          // (B,N,64)
    float* __restrict__ rsum)        // (B,N)
{
  __shared__ float xl[Tt];
  __shared__ float inv[Ss];
  __shared__ float Wl[12][16][16];   // [tile][l][col]: tiles 0-3 ig, 4-7 gg, 8-11 og
  __shared__ float bl[12][16];
  __shared__ float red[32];
  __shared__ float hds[2][64];

  const int bn = blockIdx.x;
  const int b = bn / Nn, n = bn % Nn;
  const int lane = threadIdx.x;
  const int colc = lane & 15;
  const int hi   = (lane >> 4) & 1;

  for (int i = lane; i < Tt; i += 32) xl[i] = x[(size_t)(b * Nn + n) * Tt + i];
  __syncthreads();

  float rpart = 0.f;
  for (int s = lane; s < Ss; s += 32) {
    float iv = fast_rcp(xl[Ll + s]);
    inv[s] = iv;
    rpart += iv;
  }
  // Stage W_ih tiles (ig: h4 0..63, gg: 128..191, og: 192..255) and biases.
  for (int i = lane; i < 12 * 16 * 16; i += 32) {
    int t = i >> 8, l = (i >> 4) & 15, c = i & 15;
    int h4 = (t < 4) ? (t * 16 + c)
                     : (t < 8 ? (128 + (t - 4) * 16 + c) : (192 + (t - 8) * 16 + c));
    Wl[t][l][c] = W_ih[((size_t)n * 256 + h4) * Ll + l];
  }
  for (int i = lane; i < 12 * 16; i += 32) {
    int t = i >> 4, c = i & 15;
    int h4 = (t < 4) ? (t * 16 + c)
                     : (t < 8 ? (128 + (t - 4) * 16 + c) : (192 + (t - 8) * 16 + c));
    bl[t][c] = b_ih[(size_t)n * 256 + h4] + b_hh[(size_t)n * 256 + h4];
  }
  __syncthreads();

  float hdaccs[4] = {0.f, 0.f, 0.f, 0.f};

  for (int st = 0; st < 15; ++st) {
    v2f afr[4];
#pragma unroll
    for (int ko = 0; ko < 4; ++ko) {
      int l0 = 4 * ko + 2 * hi;
      afr[ko].x = xl[st * 16 + colc + l0];
      afr[ko].y = xl[st * 16 + colc + l0 + 1];
    }
#pragma unroll
    for (int t = 0; t < 4; ++t) {
      v8f ai = {}, ag = {}, ao = {};
#pragma unroll
      for (int ko = 0; ko < 4; ++ko) {
        int l0 = 4 * ko + 2 * hi;
        v2f bi, bg, bo;
        bi.x = Wl[t][l0][colc];     bi.y = Wl[t][l0 + 1][colc];
        bg.x = Wl[4 + t][l0][colc]; bg.y = Wl[4 + t][l0 + 1][colc];
        bo.x = Wl[8 + t][l0][colc]; bo.y = Wl[8 + t][l0 + 1][colc];
        ai = wmma4(afr[ko], bi, ai);
        ag = wmma4(afr[ko], bg, ag);
        ao = wmma4(afr[ko], bo, ao);
      }
      const float biasI = bl[t][colc], biasG = bl[4 + t][colc], biasO = bl[8 + t][colc];
#pragma unroll
      for (int r = 0; r < 8; ++r) {
        int s = st * 16 + r + 8 * hi;
        float ig = ai[r] + biasI;
        float gg = ag[r] + biasG;
        float og = ao[r] + biasO;
        float cc = fast_sig(ig) * fast_tanh(gg);
        float hh = fast_sig(og) * fast_tanh(cc);
        hdaccs[t] += hh * inv[s];
      }
    }
  }
#pragma unroll
  for (int t = 0; t < 4; ++t) hds[hi][t * 16 + colc] = hdaccs[t];
  red[lane] = rpart;
  __syncthreads();
  for (int h = lane; h < 64; h += 32)
    hd[(size_t)(b * Nn + n) * 64 + h] = hds[0][h] + hds[1][h];
  if (lane == 0) {
    float s = 0.f;
    for (int i = 0; i < 32; ++i) s += red[i];
    rsum[b * Nn + n] = s;
  }
}

// ===========================================================================
// K2: per n: G[b,j,n] = (W_fc[n] @ hd[b,n] + b_fc[n,j]*r[b,n]) / S
//     WMMA: M = j (12 tiles), N = b (8 of 16 cols), K = 64.  One wave per n.
// ===========================================================================
__global__ __launch_bounds__(32) void k_graph(
    const float* __restrict__ W_fc,  // (N,N,64)
    const float* __restrict__ b_fc,  // (N,N)
    const float* __restrict__ hd,    // (B,N,64)
    const float* __restrict__ rsum,  // (B,N)
    float* __restrict__ G)           // (B,N,N)  G[b][j][n]
{
  const int n = blockIdx.x;
  const int lane = threadIdx.x;
  const int colc = lane & 15;
  const int hi   = (lane >> 4) & 1;

  v2f bbf[16];
#pragma unroll
  for (int ko = 0; ko < 16; ++ko) {
    int k0 = 4 * ko + 2 * hi;
    float bx = 0.f, by = 0.f;
    if (colc < Bb) {
      bx = hd[(size_t)(colc * Nn + n) * 64 + k0];
      by = hd[(size_t)(colc * Nn + n) * 64 + k0 + 1];
    }
    bbf[ko].x = bx; bbf[ko].y = by;
  }
  const float rs = (colc < Bb) ? rsum[colc * Nn + n] : 0.f;
  const float invS = 1.0f / (float)Ss;

  for (int jt = 0; jt < 12; ++jt) {
    const int ja = jt * 16 + colc;
    v8f acc = {};
#pragma unroll
    for (int ko = 0; ko < 16; ++ko) {
      int k0 = 4 * ko + 2 * hi;
      v2f a;
      a.x = (ja < Nn) ? W_fc[((size_t)n * Nn + ja) * 64 + k0] : 0.f;
      a.y = (ja < Nn) ? W_fc[((size_t)n * Nn + ja) * 64 + k0 + 1] : 0.f;
      acc = wmma4(a, bbf[ko], acc);
    }
    if (colc < Bb) {
#pragma unroll
      for (int r = 0; r < 8; ++r) {
        int j = jt * 16 + r + 8 * hi;
        if (j < Nn) {
          float g = (acc[r] + b_fc[(size_t)n * Nn + j] * rs) * invS;
          G[((size_t)colc * Nn + j) * Nn + n] = g;
        }
      }
    }
  }
}

// ===========================================================================
// Generic batched WMMA GEMM:  D[b, m, 0:64] (+optional relu) =
//   sum_k A[b*sAb + m*sAm + k*sAk] * Bm[b*sBb + k*64 + o]
//   D row stride ldd (writes cols 0..63 of the given pointer).
//   One wave computes one 16x64 tile.
// ===========================================================================
__global__ __launch_bounds__(32) void k_bgemm(
    const float* __restrict__ A, long sAb, int sAm, int sAk,
    const float* __restrict__ Bm, long sBb,
    float* __restrict__ D, long sDb, int ldd,
    int M, int K, int doRelu)
{
  const int b = blockIdx.y;
  const int mt = blockIdx.x;
  const int lane = threadIdx.x;
  const int colc = lane & 15;
  const int hi   = (lane >> 4) & 1;

  const float* Ab = A + (long)b * sAb;
  const float* Bp = Bm + (long)b * sBb;
  float* Db = D + (long)b * sDb;

  const int m = mt * 16 + colc;
  v8f acc0 = {}, acc1 = {}, acc2 = {}, acc3 = {};

  for (int kb = 0; kb < K; kb += 4) {
    int k0 = kb + 2 * hi;
    v2f a;
    a.x = (m < M) ? Ab[(long)m * sAm + (long)k0 * sAk] : 0.f;
    a.y = (m < M) ? Ab[(long)m * sAm + (long)(k0 + 1) * sAk] : 0.f;
    const float* Br0 = Bp + (long)k0 * 64;
    const float* Br1 = Bp + (long)(k0 + 1) * 64;
    v2f b0, b1, b2, b3;
    b0.x = Br0[colc];      b0.y = Br1[colc];
    b1.x = Br0[16 + colc]; b1.y = Br1[16 + colc];
    b2.x = Br0[32 + colc]; b2.y = Br1[32 + colc];
    b3.x = Br0[48 + colc]; b3.y = Br1[48 + colc];
    // software prefetch of next K-block (global_prefetch_b8) while WMMAs drain
    if (kb + 4 < K) {
      __builtin_prefetch(Bp + (long)(kb + 4 + 2 * hi) * 64 + colc, 0, 0);
      if (m < M) __builtin_prefetch(Ab + (long)m * sAm + (long)(kb + 4 + 2 * hi) * sAk, 0, 0);
    }
    acc0 = wmma4(a, b0, acc0);
    acc1 = wmma4(a, b1, acc1);
    acc2 = wmma4(a, b2, acc2);
    acc3 = wmma4(a, b3, acc3);
  }
#pragma unroll
  for (int r = 0; r < 8; ++r) {
    int mr = mt * 16 + r + 8 * hi;
    if (mr < M) {
      float* drow = Db + (long)mr * ldd;
      float v0 = acc0[r], v1 = acc1[r], v2 = acc2[r], v3 = acc3[r];
      if (doRelu) {
        v0 = fmaxf(v0, 0.f); v1 = fmaxf(v1, 0.f);
        v2 = fmaxf(v2, 0.f); v3 = fmaxf(v3, 0.f);
      }
      drow[colc] = v0; drow[16 + colc] = v1;
      drow[32 + colc] = v2; drow[48 + colc] = v3;
    }
  }
}

// ---------------- small glue kernels ---------------------------------------
__global__ void k_rowcol(const float* __restrict__ G, float* __restrict__ row,
                         float* __restrict__ col, int M) {
  int i = blockIdx.x * blockDim.x + threadIdx.x;
  if (i >= Bb * M) return;
  int b = i / M, m = i % M;
  const float* Gb = G + (long)b * M * M;
  float r = 0.f, c = 0.f;
  for (int k = 0; k < M; ++k) { r += Gb[(long)m * M + k]; c += Gb[(long)k * M + m]; }
  row[i] = r; col[i] = c;
}

// Hcat[:, :, 0:64] = relu(0.5*(P+Q)); P <- P/row; Q <- Q/col  (in place)
__global__ void k_out1_scale(const float* P, const float* Q, const float* row,
                             const float* col, float* Hcat, float* Pn, float* Qn,
                             int M) {
  int i = blockIdx.x * blockDim.x + threadIdx.x;
  if (i >= Bb * M * 64) return;
  int o = i & 63, bm = i >> 6;
  int b = bm / M, m = bm % M;
  float p = P[i], q = Q[i];
  Hcat[((long)b * M + m) * 192 + o] = fmaxf(0.5f * (p + q), 0.f);
  Pn[i] = p / row[bm];
  Qn[i] = q / col[bm];
}

__global__ void k_scores(const float* __restrict__ Hcat, const float* __restrict__ w,
                         float* __restrict__ scores, int M) {
  int i = blockIdx.x * blockDim.x + threadIdx.x;
  if (i >= Bb * M) return;
  float nrm = 0.f;
  for (int f = 0; f < 192; ++f) nrm += w[f] * w[f];
  nrm = sqrtf(nrm);
  const float* h = Hcat + (long)i * 192;
  float s = 0.f;
  for (int f = 0; f < 192; ++f) s += h[f] * w[f];
  scores[i] = s / nrm;
}

__global__ void k_topk(const float* __restrict__ scores, int* __restrict__ idx,
                       float* __restrict__ topv, int M, int k) {
  int b = blockIdx.x * blockDim.x + threadIdx.x;
  if (b >= Bb) return;
  bool used[Nn];
  for (int i = 0; i < M; ++i) used[i] = false;
  for (int j = 0; j < k; ++j) {
    float best = -3.0e38f; int bi = 0;
    for (int i = 0; i < M; ++i) {
      if (!used[i]) {
        float v = scores[b * M + i];
        if (v > best) { best = v; bi = i; }   // strict > : ties -> lowest index
      }
    }
    used[bi] = true;
    idx[b * k + j] = bi;
    topv[b * k + j] = best;
  }
}

__global__ void k_gather_H(const float* __restrict__ Hcat, const int* __restrict__ idx,
                           const float* __restrict__ topv, float* __restrict__ Hout,
                           int M, int k) {
  int i = blockIdx.x * blockDim.x + threadIdx.x;
  if (i >= Bb * k * 192) return;
  int f = i % 192, bj = i / 192;
  int b = bj / k, j = bj % k;
  int src = idx[b * k + j];
  float g = fast_sig(topv[b * k + j]);
  Hout[i] = Hcat[((long)b * M + src) * 192 + f] * g;
}

__global__ void k_gather_G(const float* __restrict__ G, const int* __restrict__ idx,
                           float* __restrict__ Gout, int M, int k) {
  int t = blockIdx.x * blockDim.x + threadIdx.x;
  if (t >= Bb * k * k) return;
  int j = t % k, bi = t / k;
  int b = bi / k, i = bi % k;
  Gout[t] = G[((long)b * M + idx[b * k + i]) * M + idx[b * k + j]];
}

__global__ __launch_bounds__(256) void k_final(const float* __restrict__ H2,
                                               const float* __restrict__ W_out,
                                               const float* __restrict__ b_out,
                                               float* __restrict__ out) {
  __shared__ float r0[256], r1[256];
  int b = blockIdx.x, t = threadIdx.x;
  float s0 = 0.f, s1 = 0.f;
  for (int f = t; f < 3840; f += 256) {
    float h = H2[(long)b * 3840 + f];
    s0 += h * W_out[f * 2 + 0];
    s1 += h * W_out[f * 2 + 1];
  }
  r0[t] = s0; r1[t] = s1;
  __syncthreads();
  for (int off = 128; off > 0; off >>= 1) {
    if (t < off) { r0[t] += r0[t + off]; r1[t] += r1[t + off]; }
    __syncthreads();
  }
  if (t == 0) {
    float l0 = r0[0] + b_out[0], l1 = r1[0] + b_out[1];
    float mx = fmaxf(l0, l1);
    float e0 = __expf(l0 - mx), e1 = __expf(l1 - mx);
    float inv = 1.0f / (e0 + e1);
    out[b * 2 + 0] = e0 * inv;
    out[b * 2 + 1] = e1 * inv;
  }
}

// ===========================================================================
extern "C" void kernel_launch(void* const* d_in, const int* in_sizes, int n_in,
                              void* d_out, int out_size, void* d_ws, size_t ws_size,
                              hipStream_t stream) {
  const float* x      = (const float*)d_in[0];
  const float* W_ih   = (const float*)d_in[1];
  const float* b_ih   = (const float*)d_in[2];
  const float* b_hh   = (const float*)d_in[3];
  const float* W_fc   = (const float*)d_in[4];
  const float* b_fc   = (const float*)d_in[5];
  const float* W_dgc1 = (const float*)d_in[6];
  const float* W_dgc2 = (const float*)d_in[7];
  const float* w1     = (const float*)d_in[8];
  const float* w2     = (const float*)d_in[9];
  const float* W_out  = (const float*)d_in[10];
  const float* b_out  = (const float*)d_in[11];
  float* out = (float*)d_out;

  float* p = (float*)d_ws;
  auto alloc = [&](size_t nel) { float* r = p; p += nel; return r; };

  float* hd    = alloc((size_t)Bb * Nn * 64);
  float* rsum  = alloc((size_t)Bb * Nn);
  float* G     = alloc((size_t)Bb * Nn * Nn);
  float* Ne1   = alloc((size_t)Bb * Nn * 64);
  float* P1    = alloc((size_t)Bb * Nn * 64);
  float* Q1    = alloc((size_t)Bb * Nn * 64);
  float* row1  = alloc((size_t)Bb * Nn);
  float* col1  = alloc((size_t)Bb * Nn);
  float* H1cat = alloc((size_t)Bb * Nn * 192);
  float* sc1   = alloc((size_t)Bb * Nn);
  int*   idx1  = (int*)alloc((size_t)Bb * 60);
  float* tv1   = alloc((size_t)Bb * 60);
  float* H1    = alloc((size_t)Bb * 60 * 192);
  float* G1    = alloc((size_t)Bb * 60 * 60);
  float* Ne2   = alloc((size_t)Bb * 60 * 64);
  float* P2    = alloc((size_t)Bb * 60 * 64);
  float* Q2    = alloc((size_t)Bb * 60 * 64);
  float* row2  = alloc((size_t)Bb * 60);
  float* col2  = alloc((size_t)Bb * 60);
  float* H2cat = alloc((size_t)Bb * 60 * 192);
  float* sc2   = alloc((size_t)Bb * 60);
  int*   idx2  = (int*)alloc((size_t)Bb * 20);
  float* tv2   = alloc((size_t)Bb * 20);
  float* H2    = alloc((size_t)Bb * 20 * 192);

  auto cdiv = [](int a, int b) { return (a + b - 1) / b; };

  // 1) gates GEMM + activations + reduction -> hd, rsum
  k_gates_hd<<<Bb * Nn, 32, 0, stream>>>(x, W_ih, b_ih, b_hh, hd, rsum);
  // 2) G graph
  k_graph<<<Nn, 32, 0, stream>>>(W_fc, b_fc, hd, rsum, G);

  // ---------------- level 1 dgc ----------------
  k_bgemm<<<dim3(12, Bb), 32, 0, stream>>>(x, (long)Nn * Tt, Tt, 1,
                                           W_dgc1, 0L,
                                           Ne1, (long)Nn * 64, 64, Nn, Tt, 0);
  k_rowcol<<<cdiv(Bb * Nn, 256), 256, 0, stream>>>(G, row1, col1, Nn);
  k_bgemm<<<dim3(12, Bb), 32, 0, stream>>>(G, (long)Nn * Nn, Nn, 1,
                                           Ne1, (long)Nn * 64,
                                           P1, (long)Nn * 64, 64, Nn, Nn, 0);
  k_bgemm<<<dim3(12, Bb), 32, 0, stream>>>(G, (long)Nn * Nn, 1, Nn,
                                           Ne1, (long)Nn * 64,
                                           Q1, (long)Nn * 64, 64, Nn, Nn, 0);
  k_out1_scale<<<cdiv(Bb * Nn * 64, 256), 256, 0, stream>>>(P1, Q1, row1, col1,
                                                            H1cat, P1, Q1, Nn);
  k_bgemm<<<dim3(12, Bb), 32, 0, stream>>>(G, (long)Nn * Nn, 1, Nn,   // G^T @ (P/row)
                                           P1, (long)Nn * 64,
                                           H1cat + 64, (long)Nn * 192, 192, Nn, Nn, 1);
  k_bgemm<<<dim3(12, Bb), 32, 0, stream>>>(G, (long)Nn * Nn, Nn, 1,   // G @ (Q/col)
                                           Q1, (long)Nn * 64,
                                           H1cat + 128, (long)Nn * 192, 192, Nn, Nn, 1);
  // ---------------- top-k pool 1 (k=60) ----------------
  k_scores<<<cdiv(Bb * Nn, 256), 256, 0, stream>>>(H1cat, w1, sc1, Nn);
  k_topk<<<1, Bb, 0, stream>>>(sc1, idx1, tv1, Nn, 60);
  k_gather_H<<<cdiv(Bb * 60 * 192, 256), 256, 0, stream>>>(H1cat, idx1, tv1, H1, Nn, 60);
  k_gather_G<<<cdiv(Bb * 60 * 60, 256), 256, 0, stream>>>(G, idx1, G1, Nn, 60);

  // ---------------- level 2 dgc ----------------
  k_bgemm<<<dim3(4, Bb), 32, 0, stream>>>(H1, (long)60 * 192, 192, 1,
                                          W_dgc2, 0L,
                                          Ne2, (long)60 * 64, 64, 60, 192, 0);
  k_rowcol<<<cdiv(Bb * 60, 256), 256, 0, stream>>>(G1, row2, col2, 60);
  k_bgemm<<<dim3(4, Bb), 32, 0, stream>>>(G1, (long)60 * 60, 60, 1,
                                          Ne2, (long)60 * 64,
                                          P2, (long)60 * 64, 64, 60, 60, 0);
  k_bgemm<<<dim3(4, Bb), 32, 0, stream>>>(G1, (long)60 * 60, 1, 60,
                                          Ne2, (long)60 * 64,
                                          Q2, (long)60 * 64, 64, 60, 60, 0);
  k_out1_scale<<<cdiv(Bb * 60 * 64, 256), 256, 0, stream>>>(P2, Q2, row2, col2,
                                                            H2cat, P2, Q2, 60);
  k_bgemm<<<dim3(4, Bb), 32, 0, stream>>>(G1, (long)60 * 60, 1, 60,
                                          P2, (long)60 * 64,
                                          H2cat + 64, (long)60 * 192, 192, 60, 60, 1);
  k_bgemm<<<dim3(4, Bb), 32, 0, stream>>>(G1, (long)60 * 60, 60, 1,
                                          Q2, (long)60 * 64,
                                          H2cat + 128, (long)60 * 192, 192, 60, 60, 1);
  // ---------------- top-k pool 2 (k=20) ----------------
  k_scores<<<cdiv(Bb * 60, 256), 256, 0, stream>>>(H2cat, w2, sc2, 60);
  k_topk<<<1, Bb, 0, stream>>>(sc2, idx2, tv2, 60, 20);
  k_gather_H<<<cdiv(Bb * 20 * 192, 256), 256, 0, stream>>>(H2cat, idx2, tv2, H2, 60, 20);

  // ---------------- output FC + softmax ----------------
  k_final<<<Bb, 256, 0, stream>>>(H2, W_out, b_out, out);

  (void)in_sizes; (void)n_in; (void)out_size; (void)ws_size;
}